// UncertainPointsCoordsOnGrid_22943715295832
// MI455X (gfx1250) — compile-verified
//
#include <hip/hip_runtime.h>
#include <cstdint>

// ---------------------------------------------------------------------------
// UncertainPointsCoordsOnGrid for MI455X (gfx1250, wave32).
//
// Phase 1 (memory-bound, ~159 MB @ 23.3 TB/s ~= 7us): TDM tensor_load_to_lds
//   streams 256-pixel (19456 B) tiles into LDS, double-buffered on TENSORcnt.
//   Each thread computes softmax-top2 uncertainty and an order-inverted u32
//   sort key (ascending key == descending uncertainty).
// Phase 2: 4x8-bit stable LSD radix sort of (key, index) per batch
//   (hist -> scan -> ballot-ranked scatter), ~130 MB traffic ~= 7us.
// Phase 3: emit top-65536 indices (int32) + normalized pixel-center coords.
// ---------------------------------------------------------------------------

#define B 8
#define H 512
#define W 512
#define C 19
#define NPB (H * W)              // 262144 spatial points per batch (1<<18)
#define KSEL (NPB / 4)           // 65536 selected points per batch
#define PIX 256                  // pixels per DMA tile
#define TDW (PIX * C)            // 4864 dwords per tile (19456 bytes)
#define TPB 8                    // tiles per block in phase 1
#define NTILES (B * NPB / PIX)   // 8192 total tiles
#define STILE 256                // elements per sort tile
#define STILES (NPB / STILE)     // 1024 sort tiles per batch

typedef unsigned int u32;
typedef u32 u32x4 __attribute__((ext_vector_type(4)));
typedef int i32x4 __attribute__((ext_vector_type(4)));
typedef int i32x8 __attribute__((ext_vector_type(8)));

#if __has_builtin(__builtin_amdgcn_tensor_load_to_lds)
#define USE_TDM 1
#else
#define USE_TDM 0
#endif

#if USE_TDM
// Issue a 1-D TDM DMA of TDW dwords from global `gaddr` into LDS offset
// `ldsOff`. D# layout per CDNA5 ISA section 8 (group0 128b, group1 256b).
// This toolchain (clang-23 / therock-10.0) exposes the 6-arg builtin:
//   (uint32x4 g0, int32x8 g1, int32x4 g2, int32x4 g3, int32x8 pad, i32 cpol)
__device__ __forceinline__ void tdm_issue(unsigned long long gaddr, u32 ldsOff) {
  const u32 TD0 = (u32)B * H * W * C;  // whole tensor length in dwords
  u32x4 g0;
  g0[0] = 1u;                                        // count=1, user mode
  g0[1] = ldsOff;                                    // lds_addr
  g0[2] = (u32)(gaddr & 0xFFFFFFFFull);              // global_addr[31:0]
  g0[3] = (u32)((gaddr >> 32) & 0x01FFFFFFull)       // global_addr[56:32]
          | (2u << 30);                              // type = 2 ("image")
  i32x8 g1;
  g1[0] = (int)(2u << 16);                           // data_size=2 -> 4 bytes
  g1[1] = (int)((TD0 & 0xFFFFu) << 16);              // tensor_dim0[15:0]
  g1[2] = (int)(((TD0 >> 16) & 0xFFFFu) | (1u << 16)); // dim0 hi | dim1=1 lo
  g1[3] = (int)((u32)TDW << 16);                     // dim1 hi=0 | tile_dim0
  g1[4] = 0;                                         // tile_dim1=0, tile_dim2=0
  g1[5] = (int)TD0;                                  // tensor_dim0_stride lo32
  g1[6] = 0;                                         // stride hi | dim1_stride lo
  g1[7] = 0;
  i32x4 z4 = {0, 0, 0, 0};
  i32x8 z8 = {0, 0, 0, 0, 0, 0, 0, 0};
  __builtin_amdgcn_tensor_load_to_lds(g0, g1, z4, z4, z8, 0);
}
#endif

// Descending-float -> ascending-u32 monotone key.
__device__ __forceinline__ u32 desc_key(float f) {
  u32 u = __float_as_uint(f);
  return (u & 0x80000000u) ? u : (~u & 0x7FFFFFFFu);
}

// ---------------------------------------------------------------------------
// Phase 1: uncertainty + sort keys.
// ---------------------------------------------------------------------------
__global__ __launch_bounds__(PIX) void uncert_keys(const float* __restrict__ in,
                                                   u32* __restrict__ keyOut,
                                                   u32* __restrict__ idxOut) {
  __shared__ float lbuf[2][TDW];
  const int tid = threadIdx.x;
  const int wave = tid >> 5;
  const int tile0 = blockIdx.x * TPB;

#if USE_TDM
  const unsigned long long gbase = (unsigned long long)(uintptr_t)in;
  if (wave == 0) {
    tdm_issue(gbase + (unsigned long long)tile0 * (TDW * 4),
              (u32)(uintptr_t)&lbuf[0][0]);
  }
#endif

  for (int t = 0; t < TPB; ++t) {
    const int cur = t & 1;
#if USE_TDM
    if (wave == 0) {
      if (t + 1 < TPB) {
        tdm_issue(gbase + (unsigned long long)(tile0 + t + 1) * (TDW * 4),
                  (u32)(uintptr_t)&lbuf[cur ^ 1][0]);
        __builtin_amdgcn_s_wait_tensorcnt(1);  // oldest DMA (this tile) done
      } else {
        __builtin_amdgcn_s_wait_tensorcnt(0);
      }
    }
#endif
    __syncthreads();

    const int p = (tile0 + t) * PIX + tid;     // global pixel
#if USE_TDM
    const float* v = &lbuf[cur][tid * C];      // stride 19 dwords: odd stride,
                                               // conflict-free over 64 banks
#else
    const float* v = in + (size_t)p * C;
#endif
    float x[C];
    float m1 = -3.402823e38f, m2 = -3.402823e38f;
#pragma unroll
    for (int i = 0; i < C; ++i) {
      x[i] = v[i];
      if (x[i] > m1) { m2 = m1; m1 = x[i]; }
      else if (x[i] > m2) { m2 = x[i]; }
    }
    float S = 0.0f;
#pragma unroll
    for (int i = 0; i < C; ++i) S += __expf(x[i] - m1);
    // uncertainty = p2 - p1 = (exp(m2-m1) - 1) / sum(exp(x-m1))   (<= 0)
    const float unc = (__expf(m2 - m1) - 1.0f) / S;

    keyOut[p] = desc_key(unc);
    idxOut[p] = (u32)(p & (NPB - 1));          // spatial index within batch
    __syncthreads();                            // buffer reuse fence
  }
}

// ---------------------------------------------------------------------------
// Phase 2: stable LSD radix sort, 4 passes x 8 bits, per batch.
// hist layout: hist[b][digit][tile], linear scan order == scatter base order.
// ---------------------------------------------------------------------------
__global__ __launch_bounds__(STILE) void radix_hist(const u32* __restrict__ keyIn,
                                                    u32* __restrict__ hist,
                                                    int shift) {
  const int b = blockIdx.y, t = blockIdx.x, tid = threadIdx.x;
  __shared__ u32 h[256];
  h[tid] = 0;
  __syncthreads();
  const u32 kk = keyIn[(b << 18) + (t << 8) + tid];
  atomicAdd(&h[(kk >> shift) & 255u], 1u);
  __syncthreads();
  hist[((b * 256 + tid) << 10) + t] = h[tid];
}

__global__ __launch_bounds__(1024) void radix_scan(u32* __restrict__ hist) {
  __shared__ u32 s[1024];
  __shared__ u32 carry;
  const int b = blockIdx.x, tid = threadIdx.x;
  u32* h = hist + ((size_t)b << 18);           // 256 digits * 1024 tiles
  if (tid == 0) carry = 0;
  __syncthreads();
  for (int c = 0; c < 256; ++c) {
    const u32 v = h[(c << 10) + tid];
    s[tid] = v;
    __syncthreads();
    for (int off = 1; off < 1024; off <<= 1) { // Hillis-Steele inclusive
      const u32 x = (tid >= off) ? s[tid - off] : 0u;
      __syncthreads();
      s[tid] += x;
      __syncthreads();
    }
    const u32 incl = s[tid];
    const u32 base = carry;
    h[(c << 10) + tid] = base + incl - v;      // exclusive prefix
    __syncthreads();
    if (tid == 1023) carry = base + incl;
    __syncthreads();
  }
}

__global__ __launch_bounds__(STILE) void radix_scatter(const u32* __restrict__ keyIn,
                                                       const u32* __restrict__ idxIn,
                                                       u32* __restrict__ keyOut,
                                                       u32* __restrict__ idxOut,
                                                       const u32* __restrict__ hist,
                                                       int shift) {
  const int b = blockIdx.y, t = blockIdx.x, tid = threadIdx.x;
  const int lane = tid & 31, w = tid >> 5;
  __shared__ u32 wh[8 * 256];                  // per-wave digit counts
#pragma unroll
  for (int i = 0; i < 8; ++i) wh[i * 256 + tid] = 0;
  __syncthreads();

  const int gi = (b << 18) + (t << 8) + tid;
  const u32 kk = keyIn[gi];
  const u32 ix = idxIn[gi];
  const u32 d = (kk >> shift) & 255u;

  // wave32 match-any via 8 ballots -> mask of lanes sharing my digit
  unsigned m = 0xFFFFFFFFu;
#pragma unroll
  for (int bit = 0; bit < 8; ++bit) {
    const unsigned bal = __builtin_amdgcn_ballot_w32(((d >> bit) & 1u) != 0u);
    m &= ((d >> bit) & 1u) ? bal : ~bal;
  }
  const unsigned lt = (lane == 0) ? 0u : (0xFFFFFFFFu >> (32 - lane));
  const u32 wrank = (u32)__builtin_popcount(m & lt);
  if ((unsigned)lane == (unsigned)__builtin_ctz(m))
    wh[w * 256 + d] = (u32)__builtin_popcount(m);
  __syncthreads();

  u32 prefix = 0;
  for (int w2 = 0; w2 < w; ++w2) prefix += wh[w2 * 256 + d];  // stable x-wave
  const u32 pos = hist[((b * 256 + (int)d) << 10) + t] + prefix + wrank;
  keyOut[(b << 18) + pos] = kk;
  idxOut[(b << 18) + pos] = ix;
}

// ---------------------------------------------------------------------------
// Phase 3: emit top-KSEL indices (int32 region) + (x,y) coords (float region).
// ---------------------------------------------------------------------------
__global__ __launch_bounds__(256) void emit_out(const u32* __restrict__ idxSorted,
                                                int* __restrict__ outIdx,
                                                float* __restrict__ outBase) {
  const int j = blockIdx.x * 256 + threadIdx.x;  // 0 .. B*KSEL-1
  const int b = j >> 16;                          // KSEL == 1<<16
  const int r = j & (KSEL - 1);
  const u32 ix = idxSorted[(b << 18) + r];
  outIdx[j] = (int)ix;
  float* coords = outBase + (size_t)B * KSEL;     // after index region
  coords[2 * j + 0] = (0.5f / (float)W) + (float)(ix & (W - 1)) * (1.0f / (float)W);
  coords[2 * j + 1] = (0.5f / (float)H) + (float)(ix >> 9) * (1.0f / (float)H);
}

// ---------------------------------------------------------------------------
extern "C" void kernel_launch(void* const* d_in, const int* in_sizes, int n_in,
                              void* d_out, int out_size, void* d_ws, size_t ws_size,
                              hipStream_t stream) {
  (void)in_sizes; (void)n_in; (void)out_size; (void)ws_size;
  const float* in = (const float*)d_in[0];

  // Workspace: 5 x 8 MB = 40 MB
  u32* keyA = (u32*)d_ws;
  u32* idxA = keyA + (size_t)B * NPB;
  u32* keyB = idxA + (size_t)B * NPB;
  u32* idxB = keyB + (size_t)B * NPB;
  u32* hist = idxB + (size_t)B * NPB;

  uncert_keys<<<NTILES / TPB, PIX, 0, stream>>>(in, keyA, idxA);

  u32 *ks = keyA, *is = idxA, *kd = keyB, *id = idxB;
  for (int p = 0; p < 4; ++p) {
    radix_hist<<<dim3(STILES, B), STILE, 0, stream>>>(ks, hist, p * 8);
    radix_scan<<<B, 1024, 0, stream>>>(hist);
    radix_scatter<<<dim3(STILES, B), STILE, 0, stream>>>(ks, is, kd, id, hist, p * 8);
    u32* tk = ks; ks = kd; kd = tk;
    u32* ti = is; is = id; id = ti;
  }
  // 4 passes: final sorted data back in keyA/idxA (== ks/is).

  emit_out<<<(B * KSEL) / 256, 256, 0, stream>>>(is, (int*)d_out, (float*)d_out);
}